// MLNISE_41781441856072
// MI455X (gfx1250) — compile-verified
//
#include <hip/hip_runtime.h>
#include <hip/hip_bf16.h>

// Problem constants (match reference).
#define BB 64
#define TT 500
#define NN 32
#define SAVEI 10
#define NSAVE (TT / SAVEI)          // 50
#define ICM2IFS 2.99792458e-5f
#define LP 33                       // LDS row pitch (33 floats) to avoid bank conflicts

typedef float v2f __attribute__((ext_vector_type(2)));
typedef float v8f __attribute__((ext_vector_type(8)));

// Tournament (round-robin) pairing: 16 disjoint pairs covering 0..31 each round,
// all 496 pairs visited over 31 rounds.
__device__ __forceinline__ void pairPQ(int r, int k, int& p, int& q) {
    if (k == 0) { p = 31; q = r; }
    else {
        p = (r + k) % 31;
        q = (r + 31 - k) % 31;
    }
}

// ---------------------------------------------------------------------------
// Phase 1: batched 32x32 symmetric eigendecomposition, one wave32 per matrix.
// Cyclic Jacobi, 8 sweeps x 31 rounds x 16 parallel disjoint rotations.
// A and V live in LDS (pitch 33). Single wave per matrix => lockstep lanes,
// in-order LDS within a wave, no barriers required.
// ---------------------------------------------------------------------------
__global__ __launch_bounds__(128) void eigh32_jacobi_kernel(
        const float* __restrict__ Hfull,   // [B*T, 32, 32]
        float* __restrict__ Ews,           // [B*T, 32]
        float* __restrict__ Cws)           // [B*T, 32, 32]  (row-major, cols = eigenvectors)
{
    const int wave = threadIdx.x >> 5;          // 0..3
    const int lane = threadIdx.x & 31;
    const int mat  = blockIdx.x * 4 + wave;     // exact grid: (B*T)/4 blocks

    __shared__ float Abuf[4][NN * LP];
    __shared__ float Vbuf[4][NN * LP];
    __shared__ float csb[4][32];                // c[0..15], s[16..31]

    float* A  = Abuf[wave];
    float* V  = Vbuf[wave];
    float* cs = csb[wave];

    const size_t base = (size_t)mat * (NN * NN);

    // Coalesced load of H; V = I.
    #pragma unroll
    for (int i = 0; i < NN; ++i) {
        A[i * LP + lane] = Hfull[base + i * NN + lane];
        V[i * LP + lane] = (i == lane) ? 1.0f : 0.0f;
    }

    for (int sweep = 0; sweep < 8; ++sweep) {
        for (int r = 0; r < 31; ++r) {
            // Rotation angles for the 16 pairs (lanes 0..15).
            if (lane < 16) {
                int p, q; pairPQ(r, lane, p, q);
                float app = A[p * LP + p];
                float aqq = A[q * LP + q];
                float apq = A[p * LP + q];
                float c = 1.0f, s = 0.0f;
                if (__builtin_fabsf(apq) > 1e-12f) {
                    float tau = (aqq - app) / (2.0f * apq);
                    float t   = __builtin_copysignf(1.0f, tau) /
                                (__builtin_fabsf(tau) + sqrtf(1.0f + tau * tau));
                    c = rsqrtf(1.0f + t * t);
                    s = t * c;
                }
                cs[lane]      = c;
                cs[16 + lane] = s;
            }
            // Row phase (J^T * A): lane = column j; pairs touch disjoint rows.
            #pragma unroll
            for (int k = 0; k < 16; ++k) {
                int p, q; pairPQ(r, k, p, q);
                float c = cs[k], s = cs[16 + k];
                float ap = A[p * LP + lane], aq = A[q * LP + lane];
                A[p * LP + lane] = c * ap - s * aq;
                A[q * LP + lane] = s * ap + c * aq;
            }
            // Column phase ((J^T A) * J) + eigenvector accumulation V = V*J.
            #pragma unroll
            for (int k = 0; k < 16; ++k) {
                int p, q; pairPQ(r, k, p, q);
                float c = cs[k], s = cs[16 + k];
                float ap = A[lane * LP + p], aq = A[lane * LP + q];
                A[lane * LP + p] = c * ap - s * aq;
                A[lane * LP + q] = s * ap + c * aq;
                float vp = V[lane * LP + p], vq = V[lane * LP + q];
                V[lane * LP + p] = c * vp - s * vq;
                V[lane * LP + q] = s * vp + c * vq;
            }
        }
    }

    // Eigenvalues = diagonal; eigenvectors = columns of V. No sort needed:
    // populations are invariant under column permutation/sign of C.
    Ews[(size_t)mat * NN + lane] = A[lane * LP + lane];
    #pragma unroll
    for (int i = 0; i < NN; ++i)
        Cws[base + i * NN + lane] = V[i * LP + lane];
}

// ---------------------------------------------------------------------------
// Async global->LDS streaming of one 32x32 f32 eigenvector tile (4KB) into a
// pitch-33 LDS buffer: 8 x GLOBAL_LOAD_ASYNC_TO_LDS_B128 (32 lanes x 16B each),
// tracked by ASYNCcnt. The LDS destination VGPR holds the LDS byte offset
// (flat shared address truncated to 32 bits per ISA §10.2).
// ---------------------------------------------------------------------------
__device__ __forceinline__ void async_load_C(const float* __restrict__ Cg,
                                             float* ldsbuf, int lane) {
    const unsigned lds_base = (unsigned)(size_t)(void*)ldsbuf;
    #pragma unroll
    for (int i = 0; i < 8; ++i) {
        const int c    = i * 32 + lane;        // 16B chunk id, 0..255
        const int row  = c >> 3;               // 8 chunks per 32-float row
        const int col4 = (c & 7) << 2;         // starting float within row
        unsigned           lds_addr = lds_base + (unsigned)((row * LP + col4) * 4);
        unsigned long long gaddr    = (unsigned long long)(size_t)(Cg + row * NN + col4);
        asm volatile("global_load_async_to_lds_b128 %0, %1, off"
                     :: "v"(lds_addr), "v"(gaddr) : "memory");
    }
}

__device__ __forceinline__ void wait_async0() {
    asm volatile("s_wait_asynccnt 0x0" ::: "memory");
}

// ---------------------------------------------------------------------------
// Phase 2: NISE propagation, one wave32 per batch element.
// Per step: S = C_t^T * C_{t-1} via V_WMMA_F32_16X16X4_F32 (2x2 blocks, K in
// chunks of 4 -> 32 WMMA ops), complex phiB update, per-eigenvalue phase,
// psi_site = C_t * phiB, populations saved every 10 steps.
// C tiles are triple-buffered in LDS; C_{t+1} streams in asynchronously
// (ASYNCcnt) while step t computes, removing global latency from the chain.
// ---------------------------------------------------------------------------
__global__ __launch_bounds__(32) void nise_propagate_kernel(
        const float* __restrict__ Ews,     // [B*T, 32]
        const float* __restrict__ Cws,     // [B*T, 32, 32]
        const float* __restrict__ psi0,    // [B, 32]
        const float* __restrict__ dtp,     // [B] (dt[0] used)
        float* __restrict__ out)           // [B, 50, 32]
{
    const int b    = blockIdx.x;
    const int lane = threadIdx.x;
    const int half = lane >> 4;            // 0: lanes 0-15, 1: lanes 16-31
    const int lm   = lane & 15;

    __shared__ float Cb[3][NN * LP];       // triple-buffered eigenvector matrices
    __shared__ float Sb[NN * LP];          // overlap matrix
    __shared__ float phR[NN], phI[NN], p0[NN];

    const float pc = 6.283185307179586f * ICM2IFS * dtp[0];
    const size_t cbase = (size_t)b * TT * NN * NN;
    const size_t ebase = (size_t)b * TT * NN;

    // Stream C_0 and C_1 into LDS asynchronously, then wait for both.
    async_load_C(Cws + cbase, Cb[0], lane);
    async_load_C(Cws + cbase + NN * NN, Cb[1], lane);
    wait_async0();

    // Initial eigenbasis amplitudes phiB = C0^T psi0.
    p0[lane] = psi0[b * NN + lane];
    float s0 = 0.0f;
    #pragma unroll
    for (int j = 0; j < NN; ++j)
        s0 += Cb[0][j * LP + lane] * p0[j];
    phR[lane] = s0;
    phI[lane] = 0.0f;
    out[(size_t)b * NSAVE * NN + lane] = p0[lane] * p0[lane];

    for (int t = 1; t < TT; ++t) {
        const int cur = t % 3, prv = (t - 1) % 3, nxt = (t + 1) % 3;

        // C_t must be resident (only older async loads can be outstanding;
        // async loads complete in order).
        wait_async0();
        // Kick off C_{t+1} into the third buffer; overlaps with this step.
        if (t + 1 < TT)
            async_load_C(Cws + cbase + (size_t)(t + 1) * NN * NN, Cb[nxt], lane);
        // Warm L2 two steps ahead (each lane a distinct 128B line -> 4KB).
        if (t + 2 < TT)
            __builtin_prefetch(Cws + cbase + (size_t)(t + 2) * NN * NN + lane * 32, 0, 1);

        // S = C_t^T * C_{t-1}: f32 WMMA, 2x2 16x16 blocks, K = 32 in chunks of 4.
        #pragma unroll
        for (int I = 0; I < 2; ++I) {
            #pragma unroll
            for (int J = 0; J < 2; ++J) {
                v8f acc = {};
                #pragma unroll
                for (int c4 = 0; c4 < 8; ++c4) {
                    const int k0 = 4 * c4 + 2 * half;
                    v2f a, bf;
                    // A-fragment: A[m][k] = C_t[k][m]  (transposed read from LDS)
                    a[0]  = Cb[cur][(k0 + 0) * LP + 16 * I + lm];
                    a[1]  = Cb[cur][(k0 + 1) * LP + 16 * I + lm];
                    // B-fragment: B[k][n] = C_{t-1}[k][n]
                    bf[0] = Cb[prv][(k0 + 0) * LP + 16 * J + lm];
                    bf[1] = Cb[prv][(k0 + 1) * LP + 16 * J + lm];
                    acc = __builtin_amdgcn_wmma_f32_16x16x4_f32(
                              false, a, false, bf, (short)0, acc, false, false);
                }
                #pragma unroll
                for (int rr = 0; rr < 8; ++rr)
                    Sb[(16 * I + rr + 8 * half) * LP + 16 * J + lm] = acc[rr];
            }
        }

        // phiB' = S * phiB (complex, S real).
        float tR = 0.0f, tI = 0.0f;
        #pragma unroll
        for (int j = 0; j < NN; ++j) {
            float s = Sb[lane * LP + j];
            tR += s * phR[j];
            tI += s * phI[j];
        }

        // Diagonal phase evolution: exp(-i * pc * E).
        float E  = Ews[ebase + (size_t)t * NN + lane];
        float th = pc * E, sth, cth;
        __sincosf(th, &sth, &cth);
        float nR = tR * cth + tI * sth;
        float nI = tI * cth - tR * sth;
        phR[lane] = nR;
        phI[lane] = nI;

        // psi_site = C_t * phiB'; population = |psi_site|^2.
        float aR = 0.0f, aI = 0.0f;
        #pragma unroll
        for (int j = 0; j < NN; ++j) {
            float cv = Cb[cur][lane * LP + j];
            aR += cv * phR[j];
            aI += cv * phI[j];
        }
        if ((t % SAVEI) == 0)
            out[(size_t)b * NSAVE * NN + (t / SAVEI) * NN + lane] = aR * aR + aI * aI;
    }
}

// ---------------------------------------------------------------------------
// Host launcher. Inputs: T, Er, cortim, dt, reps, psi0, Hfull.
// Workspace: E [B*T*32] + C [B*T*32*32] floats (~135.3 MB).
// ---------------------------------------------------------------------------
extern "C" void kernel_launch(void* const* d_in, const int* in_sizes, int n_in,
                              void* d_out, int out_size, void* d_ws, size_t ws_size,
                              hipStream_t stream) {
    (void)in_sizes; (void)n_in; (void)out_size; (void)ws_size;

    const float* dt    = (const float*)d_in[3];
    const float* psi0  = (const float*)d_in[5];
    const float* Hfull = (const float*)d_in[6];
    float*       out   = (float*)d_out;

    float* Ews = (float*)d_ws;                               // B*T*N floats
    float* Cws = Ews + (size_t)BB * TT * NN;                 // B*T*N*N floats

    // Phase 1: 32,000 independent eigendecompositions (4 waves/block).
    static_assert((BB * TT) % 4 == 0, "grid");
    hipLaunchKernelGGL(eigh32_jacobi_kernel,
                       dim3((BB * TT) / 4), dim3(128), 0, stream,
                       Hfull, Ews, Cws);

    // Phase 2: sequential propagation, one wave per batch element.
    hipLaunchKernelGGL(nise_propagate_kernel,
                       dim3(BB), dim3(32), 0, stream,
                       Ews, Cws, psi0, dt, out);
}